// GNN_60825326846155
// MI455X (gfx1250) — compile-verified
//
#include <hip/hip_runtime.h>

typedef __attribute__((ext_vector_type(2))) float v2f;
typedef __attribute__((ext_vector_type(8))) float v8f;

#define DIM 128
#define NCLS 40
#define LDS_PAD 132   // row pitch in floats; 132 % 64 == 4 -> conflict-free b64 reads

// ---------------------------------------------------------------------------
// Scatter-add: agg[dst[e]] += h[src[e]] for every edge. 32 lanes per edge,
// each lane owns 4 contiguous floats (128/32). fp32 L2 atomics.
// ---------------------------------------------------------------------------
__global__ __launch_bounds__(256) void scatter_kernel(
    const float* __restrict__ h, const long long* __restrict__ src,
    const long long* __restrict__ dst, float* __restrict__ agg, int E) {
  int gid = blockIdx.x * blockDim.x + threadIdx.x;
  int e = gid >> 5;
  int lane = gid & 31;
  if (e >= E) return;
  int s = (int)src[e];
  int d = (int)dst[e];
  const float4 v = *reinterpret_cast<const float4*>(h + (size_t)s * DIM + lane * 4);
  float* p = agg + (size_t)d * DIM + lane * 4;
  atomicAdd(p + 0, v.x);
  atomicAdd(p + 1, v.y);
  atomicAdd(p + 2, v.z);
  atomicAdd(p + 3, v.w);
}

// ---------------------------------------------------------------------------
// Fused dual GEMM: out = act(A1 @ B1^T + A2 @ B2^T + bias), all [*,128]x[128,128].
// Block = 256 threads (8 waves) = 16 output rows x 128 cols.
// A tiles staged coalesced into padded LDS; B read from global (L2-hot weights)
// directly in WMMA B lane layout. 64 x v_wmma_f32_16x16x4_f32 per wave.
// RELU is a template param so the epilogue is branch-free; the full-tile row
// check is hoisted to one uniform scalar branch.
// ---------------------------------------------------------------------------
template <bool RELU>
__global__ __launch_bounds__(256) void gemm_dual_kernel(
    const float* __restrict__ A1, const float* __restrict__ A2,
    const float* __restrict__ B1, const float* __restrict__ B2,
    const float* __restrict__ bias, float* __restrict__ out, int n_rows) {
  __shared__ float lA1[16 * LDS_PAD];
  __shared__ float lA2[16 * LDS_PAD];
  const int row0 = blockIdx.x * 16;
  const int t = threadIdx.x;

  // Stage both 16x128 A tiles: thread t -> row t/16, cols [8*(t%16), +8)
  {
    const int r = t >> 4;
    const int cc = (t & 15) * 8;
    const int grow = row0 + r;
    float4 v0 = make_float4(0.f, 0.f, 0.f, 0.f), v1 = v0, w0 = v0, w1 = v0;
    if (grow < n_rows) {
      const float4* p1 = reinterpret_cast<const float4*>(A1 + (size_t)grow * DIM + cc);
      const float4* p2 = reinterpret_cast<const float4*>(A2 + (size_t)grow * DIM + cc);
      v0 = p1[0]; v1 = p1[1];
      w0 = p2[0]; w1 = p2[1];
    }
    float* q1 = lA1 + r * LDS_PAD + cc;
    float* q2 = lA2 + r * LDS_PAD + cc;
    q1[0] = v0.x; q1[1] = v0.y; q1[2] = v0.z; q1[3] = v0.w;
    q1[4] = v1.x; q1[5] = v1.y; q1[6] = v1.z; q1[7] = v1.w;
    q2[0] = w0.x; q2[1] = w0.y; q2[2] = w0.z; q2[3] = w0.w;
    q2[4] = w1.x; q2[5] = w1.y; q2[6] = w1.z; q2[7] = w1.w;
  }
  __syncthreads();

  const int wave = t >> 5;
  const int lane = t & 31;
  const int half = lane >> 4;   // 0: K=k0..k0+1, 1: K=k0+2..k0+3
  const int l = lane & 15;
  const int col0 = wave * 16;

  // A operand: lane holds row M=l, v2f = A[M][k0+2*half .. +1]
  const float* arow1 = lA1 + l * LDS_PAD + 2 * half;
  const float* arow2 = lA2 + l * LDS_PAD + 2 * half;
  // B operand: lane holds col N=col0+l, v2f = W[N][k0+2*half .. +1]
  const float* brow1 = B1 + (size_t)(col0 + l) * DIM + 2 * half;
  const float* brow2 = B2 + (size_t)(col0 + l) * DIM + 2 * half;

  v8f acc = {};
#pragma unroll 4
  for (int k0 = 0; k0 < DIM; k0 += 4) {
    v2f a1 = *reinterpret_cast<const v2f*>(arow1 + k0);
    v2f bb1 = *reinterpret_cast<const v2f*>(brow1 + k0);
    acc = __builtin_amdgcn_wmma_f32_16x16x4_f32(false, a1, false, bb1,
                                                (short)0, acc, false, false);
    v2f a2 = *reinterpret_cast<const v2f*>(arow2 + k0);
    v2f bb2 = *reinterpret_cast<const v2f*>(brow2 + k0);
    acc = __builtin_amdgcn_wmma_f32_16x16x4_f32(false, a2, false, bb2,
                                                (short)0, acc, false, false);
  }

  const float bval = bias[col0 + l];
  float* obase = out + (size_t)(row0 + 8 * half) * DIM + col0 + l;
  if (row0 + 16 <= n_rows) {
    // Full tile (always taken for N % 16 == 0): straight-line stores.
#pragma unroll
    for (int r = 0; r < 8; ++r) {
      float v = acc[r] + bval;
      if (RELU) v = fmaxf(v, 0.0f);
      obase[(size_t)r * DIM] = v;
    }
  } else {
#pragma unroll
    for (int r = 0; r < 8; ++r) {
      int row = row0 + r + 8 * half;
      if (row < n_rows) {
        float v = acc[r] + bval;
        if (RELU) v = fmaxf(v, 0.0f);
        out[(size_t)row * DIM + col0 + l] = v;
      }
    }
  }
}

// ---------------------------------------------------------------------------
// Classifier: out[N,40] = H @ Wc^T + bc. Block = 96 threads (3 waves) covering
// 48 padded cols; lanes with n >= 40 contribute zero B columns and skip stores.
// ---------------------------------------------------------------------------
__global__ __launch_bounds__(96) void classifier_kernel(
    const float* __restrict__ H, const float* __restrict__ Wc,
    const float* __restrict__ bc, float* __restrict__ out, int n_rows) {
  __shared__ float lA[16 * LDS_PAD];
  const int row0 = blockIdx.x * 16;
  const int t = threadIdx.x;

  for (int i = t; i < 512; i += 96) {  // 512 float4 chunks = 16x128 tile
    int r = i >> 5;
    int cc = (i & 31) * 4;
    int grow = row0 + r;
    float4 v = make_float4(0.f, 0.f, 0.f, 0.f);
    if (grow < n_rows) v = *reinterpret_cast<const float4*>(H + (size_t)grow * DIM + cc);
    float* q = lA + r * LDS_PAD + cc;
    q[0] = v.x; q[1] = v.y; q[2] = v.z; q[3] = v.w;
  }
  __syncthreads();

  const int wave = t >> 5;
  const int lane = t & 31;
  const int half = lane >> 4;
  const int l = lane & 15;
  const int col0 = wave * 16;
  const int n = col0 + l;
  const bool valid = (n < NCLS);
  const float validf = valid ? 1.0f : 0.0f;

  const float* arow = lA + l * LDS_PAD + 2 * half;
  const float* brow = Wc + (size_t)(valid ? n : 0) * DIM + 2 * half;

  v8f acc = {};
#pragma unroll 4
  for (int k0 = 0; k0 < DIM; k0 += 4) {
    v2f a = *reinterpret_cast<const v2f*>(arow + k0);
    v2f b = *reinterpret_cast<const v2f*>(brow + k0);
    b = b * validf;  // zero out padded class columns
    acc = __builtin_amdgcn_wmma_f32_16x16x4_f32(false, a, false, b,
                                                (short)0, acc, false, false);
  }

  const float bcv = valid ? bc[n] : 0.0f;
  if (row0 + 16 <= n_rows) {
    float* obase = out + (size_t)(row0 + 8 * half) * NCLS + n;
#pragma unroll
    for (int r = 0; r < 8; ++r) {
      if (valid) obase[(size_t)r * NCLS] = acc[r] + bcv;
    }
  } else {
#pragma unroll
    for (int r = 0; r < 8; ++r) {
      int row = row0 + r + 8 * half;
      if (row < n_rows && valid) {
        out[(size_t)row * NCLS + n] = acc[r] + bcv;
      }
    }
  }
}

// ---------------------------------------------------------------------------
extern "C" void kernel_launch(void* const* d_in, const int* in_sizes, int n_in,
                              void* d_out, int out_size, void* d_ws, size_t ws_size,
                              hipStream_t stream) {
  const float* x        = (const float*)d_in[0];
  const long long* ei   = (const long long*)d_in[1];  // [2, E] int64
  const float* W1_rel   = (const float*)d_in[2];
  const float* W1_root  = (const float*)d_in[3];
  const float* b1       = (const float*)d_in[4];
  const float* W2_rel   = (const float*)d_in[5];
  const float* W2_root  = (const float*)d_in[6];
  const float* b2       = (const float*)d_in[7];
  const float* Wc       = (const float*)d_in[8];
  const float* bc       = (const float*)d_in[9];
  float* out            = (float*)d_out;

  const int N = in_sizes[0] / DIM;
  const int E = in_sizes[1] / 2;
  const long long* src = ei;
  const long long* dst = ei + E;

  // Workspace: [agg | h1], each N*128 f32 (25.6 MB). Layer-2 output reuses agg
  // (safe: each block copies its A rows to LDS + barrier before storing, and
  // A rows are block-exclusive).
  float* agg = (float*)d_ws;
  float* h1  = agg + (size_t)N * DIM;

  const size_t agg_bytes = (size_t)N * DIM * sizeof(float);
  const int mblocks = (N + 15) / 16;
  const long long sthreads_total = (long long)E * 32;
  const int sblocks = (int)((sthreads_total + 255) / 256);

  // Layer 1: agg = segsum(x); h1 = relu(agg@W1_rel^T + x@W1_root^T + b1)
  hipMemsetAsync(agg, 0, agg_bytes, stream);
  scatter_kernel<<<sblocks, 256, 0, stream>>>(x, src, dst, agg, E);
  gemm_dual_kernel<true><<<mblocks, 256, 0, stream>>>(agg, x, W1_rel, W1_root, b1, h1, N);

  // Layer 2: agg = segsum(h1); h2 = agg@W2_rel^T + h1@W2_root^T + b2 (in-place)
  hipMemsetAsync(agg, 0, agg_bytes, stream);
  scatter_kernel<<<sblocks, 256, 0, stream>>>(h1, src, dst, agg, E);
  gemm_dual_kernel<false><<<mblocks, 256, 0, stream>>>(agg, h1, W2_rel, W2_root, b2, agg, N);

  // Classifier: out = h2 @ Wc^T + bc
  classifier_kernel<<<mblocks, 96, 0, stream>>>(agg, Wc, bc, out, N);
}